// ReadinMatrix_5574867550881
// MI455X (gfx1250) — compile-verified
//
#include <hip/hip_runtime.h>

// ---------------------------------------------------------------------------
// Fused gathered low-rank projection for MI455X (gfx1250, wave32, WMMA).
//   out[b,t,a,:] = (state[b,t,a,:] @ readin[session[b]]) @ project
// ~241 MB min HBM traffic -> ~10.3us floor @ 23.3 TB/s; both GEMMs fused so
// h never leaves the WGP. True-fp32 matrix math via V_WMMA_F32_16X16X4_F32.
// B operands kept in K-pair-interleaved LDS so every B fragment is a single
// ds_load_b64 into a WMMA-ready even VGPR pair. Bulk state staging uses the
// CDNA5 async global->LDS path (ASYNCcnt) when the builtin is available.
// ---------------------------------------------------------------------------

typedef float v2f __attribute__((ext_vector_type(2)));
typedef float v8f __attribute__((ext_vector_type(8)));
typedef int   v4i __attribute__((vector_size(16)));   // matches builtin's type

typedef __attribute__((address_space(1))) v4i g_v4i;  // global (AS1)
typedef __attribute__((address_space(3))) v4i l_v4i;  // LDS (AS3)

#define B_SZ    128
#define T_SZ    512
#define A_SZ    2
#define IN_SZ   192
#define RDIM    64
#define OUT_SZ  256
#define ROWS    (T_SZ * A_SZ)        // 1024 rows per sample
#define ROWS_PER_BLOCK 128
#define WAVES_PER_BLOCK 8
#define THREADS (WAVES_PER_BLOCK * 32)
#define KCHUNK  32
#define NCHUNK  (IN_SZ / KCHUNK)     // 6

// LDS strides (floats):
//  A-side tiles ([row][K]): stride with gcd(s,64)=4 -> b64 A-frag reads hit
//    16 distinct bank-pairs per half-wave, halves disjoint (+2 words).
//  B-side tiles (K-pair interleaved [k/2][2n+(k&1)]): pair-row stride ≡ 32
//    (mod 64) -> half-wave reading pair-row 2k uses banks c..c+31, half-wave
//    reading 2k+1 uses c+32..c+63. Single ds_load_b64 per fragment.
#define SA_STRIDE 36    // state chunk   128 x 32       -> 18.0 KB
#define RD_PS     160   // readin chunk  16 pairs x 128 -> 10.0 KB
#define PJ_PS     544   // project       32 pairs x 512 -> 68.0 KB
#define H_STRIDE  68    // h staging     128 x 64       -> 34.0 KB
// total ~130 KB -> 2 blocks per 320KB WGP

#if defined(__has_builtin)
#  if __has_builtin(__builtin_amdgcn_global_load_async_to_lds_b128)
#    define HAVE_ASYNC_LDS 1
#  endif
#  if __has_builtin(__builtin_amdgcn_s_wait_asynccnt)
#    define HAVE_WAIT_ASYNC 1
#  endif
#endif

__device__ __forceinline__ void copy16_g2l(const float* g, float* l) {
#if defined(HAVE_ASYNC_LDS)
    __builtin_amdgcn_global_load_async_to_lds_b128(
        (g_v4i*)(float*)(g),    // global source (AS1), const cast away
        (l_v4i*)(l),            // LDS destination (AS3)
        /*offset=*/0, /*cpol=*/0);
#else
    *(float4*)l = *(const float4*)g;
#endif
}

__device__ __forceinline__ void wait_async0() {
#if defined(HAVE_ASYNC_LDS)
#  if defined(HAVE_WAIT_ASYNC)
    __builtin_amdgcn_s_wait_asynccnt(0);
#  else
    asm volatile("s_wait_asynccnt 0x0" ::: "memory");
#  endif
#endif
}

__global__ __launch_bounds__(THREADS)
void fused_readin_project_kernel(const float* __restrict__ state,
                                 const int*   __restrict__ session,
                                 const float* __restrict__ readin,
                                 const float* __restrict__ project,
                                 float*       __restrict__ out)
{
    __shared__ float sA [ROWS_PER_BLOCK * SA_STRIDE];
    __shared__ float sRd[(KCHUNK / 2) * RD_PS];
    __shared__ float sPj[(RDIM / 2) * PJ_PS];
    __shared__ float sH [ROWS_PER_BLOCK * H_STRIDE];

    const int tid  = threadIdx.x;
    const int lane = tid & 31;
    const int wave = tid >> 5;
    const int half = lane >> 4;   // which 16-lane half of the wave
    const int fm   = lane & 15;   // M (A/D) or N (B) index within fragment

    const int b       = blockIdx.x >> 3;   // 8 row-blocks per sample
    const int rowBase = (blockIdx.x & 7) * ROWS_PER_BLOCK;

    const float* stateB = state + ((size_t)b * ROWS + rowBase) * IN_SZ;
    const int    sess   = session[b];
    const float* rdB    = readin + (size_t)sess * IN_SZ * RDIM;
    float*       outB   = out + ((size_t)b * ROWS + rowBase) * OUT_SZ;

    // ---- stage project [64 x 256] into LDS once, K-pair interleaved:
    //      element (k, n) -> sPj[(k/2)*PJ_PS + 2n + (k&1)]
    for (int i = tid; i < (RDIM / 2) * OUT_SZ; i += THREADS) {   // 32/thread
        int p = i >> 8;          // pair row (k = 2p, 2p+1)
        int n = i & 255;
        float2 v;
        v.x = project[(2 * p)     * OUT_SZ + n];
        v.y = project[(2 * p + 1) * OUT_SZ + n];
        *(float2*)&sPj[p * PJ_PS + 2 * n] = v;
    }

    // =====================================================================
    // Stage 1: h[16x64 per wave] = stateTile[16x192] @ readin[192x64]
    // =====================================================================
    v8f hAcc[4] = {};                // 4 N-tiles of 16x16 f32 accumulators

    const int rw = wave * 16 + fm;   // A-fragment row for this lane

    for (int c = 0; c < NCHUNK; ++c) {
        __syncthreads();  // previous chunk fully consumed (also fences sPj)

        // state chunk 128x32 -> LDS via async b128 copies (ASYNCcnt path)
        for (int i = tid; i < ROWS_PER_BLOCK * (KCHUNK / 4); i += THREADS) {
            int r  = i >> 3;
            int c4 = i & 7;
            copy16_g2l(&stateB[r * IN_SZ + c * KCHUNK + c4 * 4],
                       &sA[r * SA_STRIDE + c4 * 4]);
        }
        // readin chunk 32x64 -> LDS, K-pair interleaved (coalesced reads,
        // one ds_store_b64 per pair, conflict-free banks)
        for (int i = tid; i < (KCHUNK / 2) * RDIM; i += THREADS) { // 4/thread
            int p = i >> 6;
            int n = i & 63;
            float2 v;
            v.x = rdB[(c * KCHUNK + 2 * p)     * RDIM + n];
            v.y = rdB[(c * KCHUNK + 2 * p + 1) * RDIM + n];
            *(float2*)&sRd[p * RD_PS + 2 * n] = v;
        }
        wait_async0();
        __syncthreads();

        for (int k = 0; k < KCHUNK / 4; ++k) {
            const int kr   = k * 4 + half * 2;  // this half-wave's K base
            const int prow = 2 * k + half;      // interleaved pair row
            // A fragment: lanes 0-15 hold K={0,1}, lanes 16-31 K={2,3}
            v2f a = *(const v2f*)&sA[rw * SA_STRIDE + kr];
#pragma unroll
            for (int n = 0; n < 4; ++n) {
                v2f bf = *(const v2f*)&sRd[prow * RD_PS + (n * 16 + fm) * 2];
                hAcc[n] = __builtin_amdgcn_wmma_f32_16x16x4_f32(
                    false, a, false, bf, (short)0, hAcc[n], false, false);
            }
        }
    }

    // ---- spill h (D layout) to LDS so it can be re-read in A layout.
    // D layout: VGPR r -> M = r + half*8, N = fm (per 16-column tile n).
    {
        const int rw0 = wave * 16;
#pragma unroll
        for (int n = 0; n < 4; ++n)
#pragma unroll
            for (int r = 0; r < 8; ++r)
                sH[(rw0 + r + half * 8) * H_STRIDE + n * 16 + fm] = hAcc[n][r];
    }
    // sH region is wave-private; per-wave LDS ordering + explicit dscnt wait.
    asm volatile("s_wait_dscnt 0x0" ::: "memory");

    // =====================================================================
    // Stage 2: outTile[16x256 per wave] = h[16x64] @ project[64x256]
    // =====================================================================
    v8f oAcc[16] = {};

    for (int k = 0; k < RDIM / 4; ++k) {
        const int kr   = k * 4 + half * 2;
        const int prow = 2 * k + half;
        v2f a = *(const v2f*)&sH[rw * H_STRIDE + kr];
#pragma unroll
        for (int n = 0; n < 16; ++n) {
            v2f bf = *(const v2f*)&sPj[prow * PJ_PS + (n * 16 + fm) * 2];
            oAcc[n] = __builtin_amdgcn_wmma_f32_16x16x4_f32(
                false, a, false, bf, (short)0, oAcc[n], false, false);
        }
    }

    // ---- store D-layout accumulators straight to global
    {
        const int rw0 = wave * 16;
#pragma unroll
        for (int n = 0; n < 16; ++n)
#pragma unroll
            for (int r = 0; r < 8; ++r)
                outB[(size_t)(rw0 + r + half * 8) * OUT_SZ + n * 16 + fm] =
                    oAcc[n][r];
    }
}

extern "C" void kernel_launch(void* const* d_in, const int* in_sizes, int n_in,
                              void* d_out, int out_size, void* d_ws, size_t ws_size,
                              hipStream_t stream) {
    const float* state   = (const float*)d_in[0];
    const int*   session = (const int*)  d_in[1];
    const float* readin  = (const float*)d_in[2];
    const float* project = (const float*)d_in[3];
    float*       out     = (float*)d_out;

    dim3 grid(B_SZ * (ROWS / ROWS_PER_BLOCK));  // 1024 blocks
    dim3 block(THREADS);                         // 8 wave32s
    hipLaunchKernelGGL(fused_readin_project_kernel, grid, block, 0, stream,
                       state, session, readin, project, out);
}